// Avod_79989470920814
// MI455X (gfx1250) — compile-verified
//
#include <hip/hip_runtime.h>
#include <hip/hip_bf16.h>
#include <math.h>

#define ROI 3
#define NBOX 8192
#define NPTS (NBOX * ROI * ROI)
#define NMS_SIZE 1024
#define NMS_THRESH 0.8f
#define X_MIN_C (-40.0f)
#define Z_MAX_C 70.0f
#define IMG_H 360
#define IMG_W 1200
#define BEV_H 704
#define BEV_W 800
#define NWORD 256  /* 8192 bits / 32 */

typedef __attribute__((ext_vector_type(2))) float v2f;
typedef __attribute__((ext_vector_type(8))) float v8f;

// ---------------------------------------------------------------------------
// CDNA5 async global->LDS copy (ASYNCcnt-tracked), inline asm for portability
// across ROCm 7.2 / amdgpu-toolchain. ldsAddr = LDS byte address (generic
// pointer low 32 bits), gaddr = 64-bit global address.
// ---------------------------------------------------------------------------
__device__ __forceinline__ void async_g2l_b32(unsigned ldsAddr,
                                              unsigned long long gaddr) {
    asm volatile("global_load_async_to_lds_b32 %0, %1, off"
                 :
                 : "v"(ldsAddr), "v"(gaddr)
                 : "memory");
}
__device__ __forceinline__ void wait_asynccnt0() {
    asm volatile("s_wait_asynccnt 0x0" ::: "memory");
}

// ---------------------------------------------------------------------------
// dot(pixel[32], w[32]) — pixel is 128B contiguous (channel-last layout)
// ---------------------------------------------------------------------------
__device__ __forceinline__ float dot32(const float* __restrict__ px,
                                       const float* __restrict__ w) {
    const float4* p4 = reinterpret_cast<const float4*>(px);
    const float4* w4 = reinterpret_cast<const float4*>(w);
    float acc = 0.f;
#pragma unroll
    for (int c = 0; c < 8; ++c) {
        float4 a = p4[c];
        float4 b = w4[c];
        acc += a.x * b.x + a.y * b.y + a.z * b.z + a.w * b.w;
    }
    return acc;
}

// Bilinear sample of relu(conv1x1(fmap)) at (y, x), zero outside valid range.
__device__ __forceinline__ float sampleMap(const float* __restrict__ fm,
                                           const float* __restrict__ w,
                                           float bias, int H, int W,
                                           float y, float x) {
    if (y < 0.f || y > (float)(H - 1) || x < 0.f || x > (float)(W - 1)) return 0.f;
    float y0f = floorf(y), x0f = floorf(x);
    float wy = y - y0f, wx = x - x0f;
    int y0 = min(max((int)y0f, 0), H - 1);
    int y1 = min(max((int)y0f + 1, 0), H - 1);
    int x0 = min(max((int)x0f, 0), W - 1);
    int x1 = min(max((int)x0f + 1, 0), W - 1);
    const float* r0 = fm + ((size_t)y0 * (size_t)W) * 32;
    const float* r1 = fm + ((size_t)y1 * (size_t)W) * 32;
    float v00 = fmaxf(dot32(r0 + (size_t)x0 * 32, w) + bias, 0.f);
    float v01 = fmaxf(dot32(r0 + (size_t)x1 * 32, w) + bias, 0.f);
    float v10 = fmaxf(dot32(r1 + (size_t)x0 * 32, w) + bias, 0.f);
    float v11 = fmaxf(dot32(r1 + (size_t)x1 * 32, w) + bias, 0.f);
    return v00 * (1.f - wy) * (1.f - wx) + v01 * (1.f - wy) * wx +
           v10 * wy * (1.f - wx) + v11 * wy * wx;
}

// ---------------------------------------------------------------------------
// Kernel A: fused 1x1-conv + crop_and_resize (both maps) -> feat [NBOX, 9]
// ---------------------------------------------------------------------------
__global__ void roi_feat_kernel(const float* __restrict__ img,
                                const float* __restrict__ bev,
                                const float* __restrict__ a_img,
                                const float* __restrict__ a_bev,
                                const float* __restrict__ w_img,
                                const float* __restrict__ b_img,
                                const float* __restrict__ w_bev,
                                const float* __restrict__ b_bev,
                                const float* __restrict__ img_mask,
                                const float* __restrict__ bev_mask,
                                float* __restrict__ feat) {
    int t = blockIdx.x * blockDim.x + threadIdx.x;
    if (t >= NPTS) return;
    int box = t / (ROI * ROI);
    int p = t - box * (ROI * ROI);
    int py = p / ROI, px = p - py * ROI;
    float ty = (float)py * 0.5f;
    float tx = (float)px * 0.5f;
    float mi = img_mask[0], mb = bev_mask[0];

    // boxes = anchors[:, [2,1,4,3]] = [y1, x1, y2, x2]
    float iy1 = a_img[box * 5 + 2], ix1 = a_img[box * 5 + 1];
    float iy2 = a_img[box * 5 + 4], ix2 = a_img[box * 5 + 3];
    float yi = (iy1 + (iy2 - iy1) * ty) * (float)(IMG_H - 1);
    float xi = (ix1 + (ix2 - ix1) * tx) * (float)(IMG_W - 1);
    float vi = sampleMap(img, w_img, b_img[0], IMG_H, IMG_W, yi, xi);

    float by1 = a_bev[box * 5 + 2], bx1 = a_bev[box * 5 + 1];
    float by2 = a_bev[box * 5 + 4], bx2 = a_bev[box * 5 + 3];
    float yb = (by1 + (by2 - by1) * ty) * (float)(BEV_H - 1);
    float xb = (bx1 + (bx2 - bx1) * tx) * (float)(BEV_W - 1);
    float vb = sampleMap(bev, w_bev, b_bev[0], BEV_H, BEV_W, yb, xb);

    feat[box * 9 + p] = (mi * vi + mb * vb) / (mi + mb);
}

// ---------------------------------------------------------------------------
// Kernel B: FC layer via V_WMMA_F32_16X16X4_F32.
// Block = 256 threads = 8 waves = 128 box-rows. Feat tile + padded weight
// matrix staged to LDS with async global->LDS; every WMMA fragment read is
// a branch-free LDS load so EXEC stays all-ones through the WMMAs.
// ---------------------------------------------------------------------------
__global__ void fc_wmma_kernel(const float* __restrict__ feat,
                               const float* __restrict__ w_obj,
                               const float* __restrict__ b_obj,
                               const float* __restrict__ w_off,
                               const float* __restrict__ b_off,
                               float* __restrict__ obj,
                               float* __restrict__ off) {
    __shared__ float wlds[12 * 16];    // Wpad[k][col], zero-padded
    __shared__ float flds[128 * 12];   // featPad[r][k], zero-padded
    int t = threadIdx.x;
    int rowBlock = blockIdx.x * 128;

    // stage weights: valid entries async global->LDS, pad entries zeroed
    if (t < 12 * 16) {
        int k = t / 16, col = t - (t / 16) * 16;
        if (k < 9 && col < 8) {
            const float* src = (col < 2) ? (w_obj + k * 2 + col)
                                         : (w_off + k * 6 + (col - 2));
            async_g2l_b32((unsigned)(size_t)&wlds[t],
                          (unsigned long long)(size_t)src);
        } else {
            wlds[t] = 0.f;
        }
    }
    // stage feat tile (128 rows x 9 contiguous floats) into padded [128][12]
    for (int idx = t; idx < 128 * 9; idx += 256) {
        int r = idx / 9, k = idx - r * 9;
        async_g2l_b32((unsigned)(size_t)&flds[r * 12 + k],
                      (unsigned long long)(size_t)(feat + (size_t)(rowBlock + r) * 9 + k));
    }
    for (int idx = t; idx < 128 * 3; idx += 256) {
        int r = idx / 3, k = 9 + (idx - r * 3);
        flds[r * 12 + k] = 0.f;
    }
    wait_asynccnt0();
    __syncthreads();

    int lane = t & 31;
    int waveInBlk = t >> 5;
    int sub = lane & 15;
    bool hi = lane >= 16;
    int rloc = waveInBlk * 16 + sub;   // local A-row 0..127

    v8f c = {0.f, 0.f, 0.f, 0.f, 0.f, 0.f, 0.f, 0.f};
#pragma unroll
    for (int k0 = 0; k0 < 12; k0 += 4) {
        // A 16x4: lanes 0-15 carry K=k0,k0+1; lanes 16-31 K=k0+2,k0+3
        int ka = k0 + (hi ? 2 : 0);
        v2f a;
        a.x = flds[rloc * 12 + ka];
        a.y = flds[rloc * 12 + ka + 1];
        // B 4x16: same K split, N = sub
        v2f b;
        b.x = wlds[ka * 16 + sub];
        b.y = wlds[(ka + 1) * 16 + sub];
        c = __builtin_amdgcn_wmma_f32_16x16x4_f32(false, a, false, b,
                                                  (short)0, c, false, false);
    }

    // D layout: VGPR v -> row v (lanes 0-15) / row v+8 (lanes 16-31), col = sub
    if (sub < 8) {
        float bias = (sub < 2) ? b_obj[sub] : b_off[sub - 2];
#pragma unroll
        for (int v = 0; v < 8; ++v) {
            int row = rowBlock + waveInBlk * 16 + v + (hi ? 8 : 0);
            float val = c[v] + bias;
            if (sub < 2)
                obj[row * 2 + sub] = val;
            else
                off[row * 6 + (sub - 2)] = val;
        }
    }
}

// ---------------------------------------------------------------------------
// Kernel C: decode regressed anchors, BEV boxes, scores = sigmoid(o1 - o0)
// ---------------------------------------------------------------------------
__global__ void postproc_kernel(const float* __restrict__ obj,
                                const float* __restrict__ off,
                                const float* __restrict__ fa,
                                float* __restrict__ reg,
                                float* __restrict__ bev_boxes,
                                float* __restrict__ score) {
    int i = blockIdx.x * blockDim.x + threadIdx.x;
    if (i >= NBOX) return;
    float r[6];
#pragma unroll
    for (int j = 0; j < 3; ++j)
        r[j] = fa[i * 6 + j] + off[i * 6 + j] * fa[i * 6 + 3 + j];
#pragma unroll
    for (int j = 0; j < 3; ++j)
        r[3 + j] = fa[i * 6 + 3 + j] * expf(off[i * 6 + 3 + j]);
#pragma unroll
    for (int j = 0; j < 6; ++j) reg[i * 6 + j] = r[j];

    float cx = r[0], cz = r[2], dx = r[3], dz = r[5];
    float bx1 = cx - dx * 0.5f - X_MIN_C;
    float bx2 = cx + dx * 0.5f - X_MIN_C;
    float bz1 = Z_MAX_C - (cz + dz * 0.5f);
    float bz2 = Z_MAX_C - (cz - dz * 0.5f);
    bev_boxes[i * 4 + 0] = bz1;
    bev_boxes[i * 4 + 1] = bx1;
    bev_boxes[i * 4 + 2] = bz2;
    bev_boxes[i * 4 + 3] = bx2;
    float o0 = obj[i * 2 + 0], o1 = obj[i * 2 + 1];
    score[i] = 1.f / (1.f + expf(o0 - o1));   // softmax(...)[:,1]
}

// ---------------------------------------------------------------------------
// Kernel D: stable descending argsort by rank counting (LDS-tiled O(N^2))
// ---------------------------------------------------------------------------
__global__ void rank_kernel(const float* __restrict__ score,
                            int* __restrict__ order) {
    __shared__ float stile[256];
    int i = blockIdx.x * 256 + threadIdx.x;
    float si = score[i];
    int rank = 0;
    for (int t0 = 0; t0 < NBOX; t0 += 256) {
        __syncthreads();
        stile[threadIdx.x] = score[t0 + threadIdx.x];
        __syncthreads();
#pragma unroll 8
        for (int u = 0; u < 256; ++u) {
            int j = t0 + u;
            float sj = stile[u];
            rank += ((sj > si) || (sj == si && j < i)) ? 1 : 0;
        }
    }
    order[rank] = i;   // total order with index tie-break -> permutation
}

// Gather boxes into sorted order + precompute areas.
__global__ void gather_sorted_kernel(const int* __restrict__ order,
                                     const float* __restrict__ bev_boxes,
                                     float* __restrict__ sbox,
                                     float* __restrict__ sarea) {
    int i = blockIdx.x * blockDim.x + threadIdx.x;
    if (i >= NBOX) return;
    int idx = order[i];
    float y1 = bev_boxes[idx * 4 + 0], x1 = bev_boxes[idx * 4 + 1];
    float y2 = bev_boxes[idx * 4 + 2], x2 = bev_boxes[idx * 4 + 3];
    sbox[i * 4 + 0] = y1;
    sbox[i * 4 + 1] = x1;
    sbox[i * 4 + 2] = y2;
    sbox[i * 4 + 3] = x2;
    sarea[i] = (y2 - y1) * (x2 - x1);
}

// ---------------------------------------------------------------------------
// Kernel E: suppression bitmask rows. block = row i, thread = 32-j word.
// ---------------------------------------------------------------------------
__global__ void iou_mask_kernel(const float* __restrict__ sbox,
                                const float* __restrict__ sarea,
                                unsigned* __restrict__ mask) {
    int i = blockIdx.x;
    int jblk = threadIdx.x;
    float y1 = sbox[i * 4 + 0], x1 = sbox[i * 4 + 1];
    float y2 = sbox[i * 4 + 2], x2 = sbox[i * 4 + 3];
    float ai = sarea[i];
    unsigned bits = 0u;
    int jbase = jblk * 32;
#pragma unroll 4
    for (int u = 0; u < 32; ++u) {
        int j = jbase + u;
        float by1 = sbox[j * 4 + 0], bx1 = sbox[j * 4 + 1];
        float by2 = sbox[j * 4 + 2], bx2 = sbox[j * 4 + 3];
        float ty = fmaxf(y1, by1), tx = fmaxf(x1, bx1);
        float ry = fminf(y2, by2), rx = fminf(x2, bx2);
        float h = fmaxf(ry - ty, 0.f), w = fmaxf(rx - tx, 0.f);
        float inter = h * w;
        float iou = inter / (ai + sarea[j] - inter + 1e-9f);
        bits |= (iou > NMS_THRESH) ? (1u << u) : 0u;
    }
    mask[(size_t)i * NWORD + jblk] = bits;
}

// ---------------------------------------------------------------------------
// Kernel F: greedy scan on a SINGLE wave32. Suppression bitmask lives in
// registers (8 words/lane, statically indexed); keep-bit broadcast via
// wave-lockstep __shfl -> zero barriers; next mask row prefetched.
// ---------------------------------------------------------------------------
__global__ void nms_scan_kernel(const unsigned* __restrict__ mask,
                                int* __restrict__ keep) {
    int lane = threadIdx.x;        // launched with 32 threads (one wave)
    unsigned supp[8] = {0u, 0u, 0u, 0u, 0u, 0u, 0u, 0u};
#pragma unroll
    for (int slot = 0; slot < 8; ++slot) {     // supp[slot]: words slot*32+lane
#pragma unroll 1
        for (int iw = 0; iw < 32; ++iw) {      // word = slot*32 + iw
#pragma unroll 1
            for (int b = 0; b < 32; ++b) {
                int i = (slot << 10) | (iw << 5) | b;
                // broadcast current suppression word for box i from lane iw
                unsigned wv = (unsigned)__shfl((int)supp[slot], iw, 32);
                int k = ((wv >> b) & 1u) ? 0 : 1;
                if (i + 1 < NBOX)
                    __builtin_prefetch(mask + (size_t)(i + 1) * NWORD + lane * 8, 0, 3);
                if (k) {
                    const unsigned* row = mask + (size_t)i * NWORD;
#pragma unroll
                    for (int s = 0; s < 8; ++s) supp[s] |= row[s * 32 + lane];
                }
                if (lane == 0) keep[i] = k;
            }
        }
    }
}

// ---------------------------------------------------------------------------
// Kernel G: selection positions (kept-in-order first, then suppressed) and
// gather reg_anchors / objectness into d_out.
// ---------------------------------------------------------------------------
__global__ void select_gather_kernel(const int* __restrict__ keep,
                                     const int* __restrict__ order,
                                     const float* __restrict__ reg,
                                     const float* __restrict__ obj,
                                     float* __restrict__ out) {
    __shared__ int pk[256];
    int t = threadIdx.x;
    int base = t * 32;
    int ck = 0;
    for (int u = 0; u < 32; ++u) ck += keep[base + u];
    pk[t] = ck;
    __syncthreads();
    for (int d = 1; d < 256; d <<= 1) {        // Hillis-Steele inclusive scan
        int val = (t >= d) ? pk[t - d] : 0;
        __syncthreads();
        pk[t] += val;
        __syncthreads();
    }
    int nKept = pk[255];
    int kr = (t == 0) ? 0 : pk[t - 1];   // exclusive kept-prefix
    int sr = base - kr;                  // exclusive suppressed-prefix
    for (int u = 0; u < 32; ++u) {
        int i = base + u;
        int k = keep[i];
        int pos = k ? kr : (nKept + sr);
        if (pos < NMS_SIZE) {
            int idx = order[i];
#pragma unroll
            for (int c = 0; c < 6; ++c) out[pos * 6 + c] = reg[idx * 6 + c];
            out[NMS_SIZE * 6 + pos * 2 + 0] = obj[idx * 2 + 0];
            out[NMS_SIZE * 6 + pos * 2 + 1] = obj[idx * 2 + 1];
        }
        kr += k;
        sr += 1 - k;
    }
}

// ---------------------------------------------------------------------------
// Launch
// ---------------------------------------------------------------------------
extern "C" void kernel_launch(void* const* d_in, const int* in_sizes, int n_in,
                              void* d_out, int out_size, void* d_ws, size_t ws_size,
                              hipStream_t stream) {
    const float* img_map  = (const float*)d_in[0];
    const float* bev_map  = (const float*)d_in[1];
    const float* a_img    = (const float*)d_in[2];
    const float* a_bev    = (const float*)d_in[3];
    const float* fanch    = (const float*)d_in[4];
    const float* w_img    = (const float*)d_in[5];
    const float* b_img    = (const float*)d_in[6];
    const float* w_bev    = (const float*)d_in[7];
    const float* b_bev    = (const float*)d_in[8];
    const float* w_obj    = (const float*)d_in[9];
    const float* b_obj    = (const float*)d_in[10];
    const float* w_off    = (const float*)d_in[11];
    const float* b_off    = (const float*)d_in[12];
    const float* img_mask = (const float*)d_in[13];
    const float* bev_mask = (const float*)d_in[14];
    float* out = (float*)d_out;

    char* ws = (char*)d_ws;
    size_t o = 0;
    auto alloc = [&](size_t bytes) {
        char* p = ws + o;
        o = (o + bytes + 255) & ~(size_t)255;
        return p;
    };
    float*    feat  = (float*)alloc((size_t)NBOX * 9 * 4);
    float*    obj   = (float*)alloc((size_t)NBOX * 2 * 4);
    float*    off   = (float*)alloc((size_t)NBOX * 6 * 4);
    float*    reg   = (float*)alloc((size_t)NBOX * 6 * 4);
    float*    bevb  = (float*)alloc((size_t)NBOX * 4 * 4);
    float*    score = (float*)alloc((size_t)NBOX * 4);
    int*      order = (int*)alloc((size_t)NBOX * 4);
    float*    sbox  = (float*)alloc((size_t)NBOX * 4 * 4);
    float*    sarea = (float*)alloc((size_t)NBOX * 4);
    int*      keep  = (int*)alloc((size_t)NBOX * 4);
    unsigned* mask  = (unsigned*)alloc((size_t)NBOX * NWORD * 4);
    (void)ws_size; (void)n_in; (void)in_sizes; (void)out_size;

    roi_feat_kernel<<<(NPTS + 255) / 256, 256, 0, stream>>>(
        img_map, bev_map, a_img, a_bev, w_img, b_img, w_bev, b_bev,
        img_mask, bev_mask, feat);

    fc_wmma_kernel<<<NBOX / 128, 256, 0, stream>>>(
        feat, w_obj, b_obj, w_off, b_off, obj, off);

    postproc_kernel<<<NBOX / 256, 256, 0, stream>>>(obj, off, fanch, reg, bevb, score);

    rank_kernel<<<NBOX / 256, 256, 0, stream>>>(score, order);

    gather_sorted_kernel<<<NBOX / 256, 256, 0, stream>>>(order, bevb, sbox, sarea);

    iou_mask_kernel<<<NBOX, NWORD, 0, stream>>>(sbox, sarea, mask);

    nms_scan_kernel<<<1, 32, 0, stream>>>(mask, keep);

    select_gather_kernel<<<1, 256, 0, stream>>>(keep, order, reg, obj, out);
}